// ODEEncoder_62551903699020
// MI455X (gfx1250) — compile-verified
//
#include <hip/hip_runtime.h>
#include <stdint.h>

#define N_USERS 200000
#define N_ITEMS 100000
#define N_NODES (N_USERS + N_ITEMS)
#define N_EDGES 4800000
#define EMB     64
#define BATCH   4096
#define NSEL    (2 * BATCH)          // 8192 selected output rows

#define THREADS 256
#define WAVES_PER_BLOCK (THREADS / 32)
#define EDGE_BLOCKS 1024             // 8192 wave32 waves over 150000 chunks

#define NBM_WORDS ((N_NODES + 31) / 32)                              // 9375
#define NBM_PAD   (((NBM_WORDS + THREADS - 1) / THREADS) * THREADS)  // 9472

// ---- CDNA5 async global->LDS path (guarded; falls back to plain loads) ----
#if defined(__AMDGCN__) && defined(__has_builtin)
#if __has_builtin(__builtin_amdgcn_global_load_async_to_lds_b32) && \
    __has_builtin(__builtin_amdgcn_s_wait_asynccnt)
#define USE_ASYNC_LDS 1
#endif
#endif
#ifndef USE_ASYNC_LDS
#define USE_ASYNC_LDS 0
#endif

#if USE_ASYNC_LDS
typedef __attribute__((address_space(1))) int as1_int;   // global
typedef __attribute__((address_space(3))) int as3_int;   // LDS
__device__ __forceinline__ void async_b32(const void* gptr, void* lptr) {
  // one b32 per lane: global -> LDS, tracked by ASYNCcnt
  __builtin_amdgcn_global_load_async_to_lds_b32(
      (as1_int*)(uintptr_t)gptr,
      (as3_int*)(uint32_t)(uintptr_t)lptr,
      0, 0);
}
#endif

// ---------------- kernel 0: init workspace ----------------
__global__ void k_init(int* __restrict__ map, unsigned* __restrict__ bitmap,
                       float* __restrict__ acc) {
  int i = blockIdx.x * blockDim.x + threadIdx.x;
  if (i < NSEL * EMB) acc[i] = 0.0f;
  if (i < N_NODES)    map[i] = -1;
  if (i < NBM_PAD)    bitmap[i] = 0u;
}

// ---------------- kernel 1: node -> slot map + membership bitmap ----------------
__global__ void k_build_map(const int* __restrict__ uid,
                            const int* __restrict__ iid,
                            int* __restrict__ map,
                            unsigned* __restrict__ bitmap) {
  int j = blockIdx.x * blockDim.x + threadIdx.x;
  if (j >= NSEL) return;
  int node = (j < BATCH) ? uid[j] : (N_USERS + iid[j - BATCH]);
  map[node] = j;   // duplicates: any representative slot is correct
  atomicOr(bitmap + (node >> 5), 1u << (node & 31));
}

// ---------------- kernel 2: bitmap-filtered, ballot-compacted SpMM ----------------
__global__ void __launch_bounds__(THREADS)
k_spmm_edges(const int*      __restrict__ rows,
             const int*      __restrict__ cols,
             const float*    __restrict__ vals,
             const float*    __restrict__ uemb,
             const float*    __restrict__ iemb,
             const int*      __restrict__ map,
             const unsigned* __restrict__ bitmap,
             float*          __restrict__ acc) {
  const int lane = threadIdx.x & 31;
  const int wl   = threadIdx.x >> 5;
  const long long wid = (long long)blockIdx.x * WAVES_PER_BLOCK + wl;
  const long long nw  = (long long)gridDim.x * WAVES_PER_BLOCK;
  const long long nchunks = N_EDGES / 32;   // 150000, exact

  __shared__ unsigned sbm[NBM_PAD];                      // 37 KB membership bitmap
  __shared__ int      srows[WAVES_PER_BLOCK][2][32];     // 2 KB row double-buffer

  // Stage the bitmap into LDS (sequential, full-line utilization).
  // NBM_PAD is a multiple of THREADS -> uniform trip count, full EXEC.
#if USE_ASYNC_LDS
  for (int w = threadIdx.x; w < NBM_PAD; w += THREADS)
    async_b32(bitmap + w, &sbm[w]);
  __builtin_amdgcn_s_wait_asynccnt(0);
#else
  for (int w = threadIdx.x; w < NBM_PAD; w += THREADS)
    sbm[w] = bitmap[w];
#endif
  __syncthreads();

#if USE_ASYNC_LDS
  if (wid < nchunks)
    async_b32(rows + wid * 32 + lane, &srows[wl][0][lane]);
#endif

  int buf = 0;
  for (long long c = wid; c < nchunks; c += nw) {
    const long long cn = c + nw;
    int r;
#if USE_ASYNC_LDS
    if (cn < nchunks) {
      async_b32(rows + cn * 32 + lane, &srows[wl][buf ^ 1][lane]);
      __builtin_amdgcn_s_wait_asynccnt(1);   // chunk c landed in LDS
    } else {
      __builtin_amdgcn_s_wait_asynccnt(0);
    }
    r = srows[wl][buf][lane];
#else
    if (cn < nchunks) __builtin_prefetch(rows + cn * 32 + lane, 0, 1);
    r = rows[c * 32 + lane];
#endif
    // LDS bitmap test instead of random 4B L2 gather (saves ~600MB of line fills)
    unsigned wbits = sbm[r >> 5];
    bool hit = (wbits >> (r & 31)) & 1u;
    int slot = hit ? map[r] : -1;            // L2 gather only on ~2.7% of edges
    unsigned long long m = __ballot(hit);
    while (m) {
      int src = __ffsll((long long)m) - 1;
      m &= m - 1;
      int   es = (int)(c * 32) + src;
      int   sl = __shfl(slot, src, 32);
      float v  = vals[es];
      int   cc = cols[es];
      const float* xr = (cc < N_USERS)
                          ? (uemb + (long long)cc * EMB)
                          : (iemb + (long long)(cc - N_USERS) * EMB);
      // 32 lanes cooperate: one b64 (float2) each = coalesced 256B row gather
      float2 ab = ((const float2*)xr)[lane];
      float* dst = acc + (long long)sl * EMB + 2 * lane;
      atomicAdd(dst,     v * ab.x);
      atomicAdd(dst + 1, v * ab.y);
    }
    buf ^= 1;
  }
}

// ---------------- kernel 3: out = 2*x0[node] + acc[map[node]] ----------------
__global__ void k_gather_out(const int*   __restrict__ uid,
                             const int*   __restrict__ iid,
                             const int*   __restrict__ map,
                             const float* __restrict__ acc,
                             const float* __restrict__ uemb,
                             const float* __restrict__ iemb,
                             float*       __restrict__ out) {
  int idx = blockIdx.x * blockDim.x + threadIdx.x;
  if (idx >= NSEL * EMB) return;
  int j = idx >> 6;
  int k = idx & 63;
  int node = (j < BATCH) ? uid[j] : (N_USERS + iid[j - BATCH]);
  int rep  = map[node];
  const float* xr = (node < N_USERS)
                      ? (uemb + (long long)node * EMB)
                      : (iemb + (long long)(node - N_USERS) * EMB);
  out[idx] = 2.0f * xr[k] + acc[(long long)rep * EMB + k];
}

// ---------------- launcher ----------------
extern "C" void kernel_launch(void* const* d_in, const int* in_sizes, int n_in,
                              void* d_out, int out_size, void* d_ws, size_t ws_size,
                              hipStream_t stream) {
  const float* uemb  = (const float*)d_in[0];
  const float* iemb  = (const float*)d_in[1];
  const int*   arow  = (const int*)  d_in[2];
  const int*   acol  = (const int*)  d_in[3];
  const float* avals = (const float*)d_in[4];
  const int*   uid   = (const int*)  d_in[5];
  const int*   iid   = (const int*)  d_in[6];
  float* out = (float*)d_out;

  // workspace: map[N_NODES] ints | bitmap[NBM_PAD] u32 | acc[NSEL*EMB] floats
  char* ws = (char*)d_ws;
  int* map = (int*)ws;
  size_t off = ((size_t)N_NODES * sizeof(int) + 255) & ~(size_t)255;
  unsigned* bitmap = (unsigned*)(ws + off);
  off += ((size_t)NBM_PAD * sizeof(unsigned) + 255) & ~(size_t)255;
  float* acc = (float*)(ws + off);

  int init_n = NSEL * EMB;  // 524288 >= N_NODES >= NBM_PAD
  k_init<<<(init_n + THREADS - 1) / THREADS, THREADS, 0, stream>>>(map, bitmap, acc);
  k_build_map<<<(NSEL + THREADS - 1) / THREADS, THREADS, 0, stream>>>(uid, iid, map, bitmap);
  k_spmm_edges<<<EDGE_BLOCKS, THREADS, 0, stream>>>(arow, acol, avals,
                                                    uemb, iemb, map, bitmap, acc);
  k_gather_out<<<(NSEL * EMB) / THREADS, THREADS, 0, stream>>>(uid, iid, map, acc,
                                                               uemb, iemb, out);
}